// Decoder_3264175145283
// MI455X (gfx1250) — compile-verified
//
#include <hip/hip_runtime.h>
#include <math.h>

#define BB 64
#define SS 1024
#define HH 512
#define OUTN 256
#define STEPSN 32
#define KX (OUTN + HH)      // 768  (dec_in | context)
#define KTOT (KX + HH)      // 1280 (dec_in | context | h)
#define G4 (4 * HH)         // 2048 gate width
#define KC 64               // GEMM K-chunk staged in LDS
#define LDSROW 68           // padded LDS row stride (floats): 16B-aligned rows AND
                            // conflict-free fragment reads (bank = 4*m + k mod 64)
#define SCH 16              // context s-chunks
#define SLEN (SS / SCH)     // 64

typedef __attribute__((ext_vector_type(2))) float v2f;
typedef __attribute__((ext_vector_type(8))) float v8f;

__device__ __forceinline__ float wave_sum(float v) {
#pragma unroll
  for (int m = 16; m > 0; m >>= 1) v += __shfl_xor(v, m, 32);
  return v;
}
__device__ __forceinline__ float wave_max(float v) {
#pragma unroll
  for (int m = 16; m > 0; m >>= 1) v = fmaxf(v, __shfl_xor(v, m, 32));
  return v;
}
// 256-thread (8-wave) block reductions
__device__ __forceinline__ float block_sum(float v, float* red) {
  int lane = threadIdx.x & 31, wid = threadIdx.x >> 5;
  v = wave_sum(v);
  __syncthreads();
  if (lane == 0) red[wid] = v;
  __syncthreads();
  if (wid == 0) {
    float r = (lane < 8) ? red[lane] : 0.f;
    r = wave_sum(r);
    if (lane == 0) red[0] = r;
  }
  __syncthreads();
  return red[0];
}
__device__ __forceinline__ float block_max(float v, float* red) {
  int lane = threadIdx.x & 31, wid = threadIdx.x >> 5;
  v = wave_max(v);
  __syncthreads();
  if (lane == 0) red[wid] = v;
  __syncthreads();
  if (wid == 0) {
    float r = (lane < 8) ? red[lane] : -INFINITY;
    r = wave_max(r);
    if (lane == 0) red[0] = r;
  }
  __syncthreads();
  return red[0];
}

// --- one-time: enc_energy[b,s] = enc[b,s,:]·wa_e + attn_b  (one wave per row)
__global__ void enc_energy_kernel(const float* __restrict__ enc,
                                  const float* __restrict__ attn_w,
                                  const float* __restrict__ attn_b,
                                  float* __restrict__ ee) {
  int wid = threadIdx.x >> 5, lane = threadIdx.x & 31;
  int row = blockIdx.x * 8 + wid;                 // row in [0, B*S)
  const float4* p = (const float4*)(enc + (size_t)row * HH) + lane;
  const float4* wv = (const float4*)attn_w + lane;
  float acc = 0.f;
#pragma unroll
  for (int j = 0; j < HH / 128; ++j) {            // 4 iters of float4
    float4 e = p[j * 32];
    float4 a = wv[j * 32];
    acc = fmaf(e.x, a.x, acc);
    acc = fmaf(e.y, a.y, acc);
    acc = fmaf(e.z, a.z, acc);
    acc = fmaf(e.w, a.w, acc);
  }
  acc = wave_sum(acc);
  if (lane == 0) ee[row] = acc + attn_b[0];
}

// --- one-time: init xh (dec0=0, h=hidden), c=0, bias=b_ih+b_hh
__global__ void init_kernel(const float* __restrict__ hidden,
                            const float* __restrict__ b_ih,
                            const float* __restrict__ b_hh,
                            float* __restrict__ xh, float* __restrict__ c,
                            float* __restrict__ bias) {
  int idx = blockIdx.x * 256 + threadIdx.x;       // over B*KTOT
  if (idx < BB * KTOT) {
    int b = idx / KTOT, k = idx % KTOT;
    if (k >= KX) {
      xh[idx] = hidden[b * HH + (k - KX)];
      c[b * HH + (k - KX)] = 0.f;
    } else {
      xh[idx] = 0.f;                              // dec0 + context slot
    }
  }
  if (idx < G4) bias[idx] = b_ih[idx] + b_hh[idx];
}

// --- per step: energy = ee + h·wa_d ; softmax over S  (one block per batch)
__global__ void attn_softmax_kernel(const float* __restrict__ ee,
                                    const float* __restrict__ attn_w,
                                    const float* __restrict__ xh,
                                    float* __restrict__ w) {
  __shared__ float red[8];
  int b = blockIdx.x, t = threadIdx.x;            // 256 threads
  const float* wa_d = attn_w + HH;
  const float* hb = xh + b * KTOT + KX;
  float p = fmaf(hb[t], wa_d[t], hb[t + 256] * wa_d[t + 256]);
  float hdot = block_sum(p, red);

  float e[4];
  float m = -INFINITY;
#pragma unroll
  for (int j = 0; j < 4; ++j) {
    e[j] = ee[b * SS + t + 256 * j] + hdot;
    m = fmaxf(m, e[j]);
  }
  m = block_max(m, red);
  float s = 0.f;
#pragma unroll
  for (int j = 0; j < 4; ++j) { e[j] = expf(e[j] - m); s += e[j]; }
  s = block_sum(s, red);
  float inv = 1.f / s;
#pragma unroll
  for (int j = 0; j < 4; ++j) w[b * SS + t + 256 * j] = e[j] * inv;
}

// --- per step, phase 1: partial context over an s-chunk (float4 / b128 path)
// grid (SCH, BB) x 128 threads -> 1024 blocks for L2-bandwidth occupancy
__global__ void context_partial_kernel(const float* __restrict__ enc,
                                       const float* __restrict__ w,
                                       float* __restrict__ part) {
  int ch = blockIdx.x, b = blockIdx.y, t = threadIdx.x;  // 128 threads, h = 4t..4t+3
  int s0 = ch * SLEN;
  const float4* eb = (const float4*)(enc + ((size_t)b * SS + s0) * HH) + t;
  const float* wb = w + b * SS + s0;
  float4 acc = make_float4(0.f, 0.f, 0.f, 0.f);
#pragma unroll 4
  for (int s = 0; s < SLEN; ++s) {
    float4 e = eb[(size_t)s * (HH / 4)];
    float ws = wb[s];
    acc.x = fmaf(ws, e.x, acc.x);
    acc.y = fmaf(ws, e.y, acc.y);
    acc.z = fmaf(ws, e.z, acc.z);
    acc.w = fmaf(ws, e.w, acc.w);
  }
  ((float4*)(part + ((size_t)ch * BB + b) * HH))[t] = acc;
}

// --- per step, phase 2: reduce partials into xh's context slot (float4)
__global__ void context_reduce_kernel(const float* __restrict__ part,
                                      float* __restrict__ xh) {
  int idx = blockIdx.x * 256 + threadIdx.x;       // B*H/4 = 8192
  int b = idx / (HH / 4), t = idx % (HH / 4);
  float4 acc = make_float4(0.f, 0.f, 0.f, 0.f);
#pragma unroll
  for (int ch = 0; ch < SCH; ++ch) {
    float4 p = ((const float4*)(part + ((size_t)ch * BB + b) * HH))[t];
    acc.x += p.x; acc.y += p.y; acc.z += p.z; acc.w += p.w;
  }
  ((float4*)(xh + (size_t)b * KTOT + OUTN))[t] = acc;
}

// --- fp32 WMMA GEMM with LDS tiling.
// Block = 512 threads = 16 waves; computes a 64(M) x 64(N) macro-tile.
// Wave w: m-tile = w&3, n-tile = w>>2. K staged in 64-wide LDS panels via b128.
// C[m,n] = sum_k A1[m,k]W1[n,k] (+ sum_k A2[m,k]W2[n,k]) + bias[n]
__global__ void gemm_wmma_tiled(const float* __restrict__ A1, const float* __restrict__ W1, int K1,
                                const float* __restrict__ A2, const float* __restrict__ W2, int K2,
                                int lda, const float* __restrict__ bias,
                                float* __restrict__ C0, int ldc0,
                                float* __restrict__ C1, int ldc1) {
  __shared__ float As[64 * LDSROW];
  __shared__ float Bs[64 * LDSROW];
  int tid = threadIdx.x;
  int lane = tid & 31, wave = tid >> 5;
  int mt = wave & 3, nt = wave >> 2;
  int half = lane >> 4, l = lane & 15;
  int n0 = blockIdx.x * 64;

  v8f acc = {};
  for (int seg = 0; seg < 2; ++seg) {
    const float* A = seg ? A2 : A1;
    const float* W = seg ? W2 : W1;
    int K = seg ? K2 : K1;
    if (K == 0) break;
    for (int kb = 0; kb < K; kb += KC) {
      __syncthreads();
      // stage 64x64 panels with float4 (global_load_b128 / ds_store_b128)
#pragma unroll
      for (int i = tid; i < 64 * (KC / 4); i += 512) {  // 2 iters
        int r = i >> 4, c4 = (i & 15) << 2;             // 16 float4 per row
        float4 av = *(const float4*)(A + (size_t)r * lda + kb + c4);
        float4 wv = *(const float4*)(W + (size_t)(n0 + r) * K + kb + c4);
        *(float4*)(&As[r * LDSROW + c4]) = av;
        *(float4*)(&Bs[r * LDSROW + c4]) = wv;
      }
      __syncthreads();
      const float* ap = &As[(mt * 16 + l) * LDSROW + 2 * half];
      const float* bp = &Bs[(nt * 16 + l) * LDSROW + 2 * half];
#pragma unroll
      for (int kk = 0; kk < KC; kk += 4) {
        v2f av = *(const v2f*)(ap + kk);
        v2f bv = *(const v2f*)(bp + kk);
        acc = __builtin_amdgcn_wmma_f32_16x16x4_f32(false, av, false, bv,
                                                    (short)0, acc, false, false);
      }
    }
  }
  int col = n0 + nt * 16 + l;
  float bvs = bias[col];
#pragma unroll
  for (int r = 0; r < 8; ++r) {
    int row = mt * 16 + r + 8 * half;               // C layout: VGPR r -> M=r / M=r+8
    float val = acc[r] + bvs;
    C0[(size_t)row * ldc0 + col] = val;
    if (C1) C1[(size_t)row * ldc1 + col] = val;
  }
}

// --- per step: LSTM elementwise; writes h_new into xh's h-slot
__global__ void lstm_kernel(const float* __restrict__ gates,
                            float* __restrict__ c, float* __restrict__ xh) {
  int idx = blockIdx.x * 256 + threadIdx.x;       // B*H
  int b = idx >> 9, t = idx & 511;
  const float* g = gates + b * G4;
  float iv = g[t], fv = g[HH + t], gv = g[2 * HH + t], ov = g[3 * HH + t];
  float si = 1.f / (1.f + expf(-iv));
  float sf = 1.f / (1.f + expf(-fv));
  float so = 1.f / (1.f + expf(-ov));
  float cn = fmaf(sf, c[idx], si * tanhf(gv));
  float hn = so * tanhf(cn);
  c[idx] = cn;
  xh[b * KTOT + KX + t] = hn;
}

extern "C" void kernel_launch(void* const* d_in, const int* in_sizes, int n_in,
                              void* d_out, int out_size, void* d_ws, size_t ws_size,
                              hipStream_t stream) {
  const float* enc    = (const float*)d_in[0];
  const float* hidden = (const float*)d_in[1];
  const float* attn_w = (const float*)d_in[2];
  const float* attn_b = (const float*)d_in[3];
  const float* w_ih   = (const float*)d_in[4];
  const float* w_hh   = (const float*)d_in[5];
  const float* b_ih   = (const float*)d_in[6];
  const float* b_hh   = (const float*)d_in[7];
  const float* fc_w   = (const float*)d_in[8];
  const float* fc_b   = (const float*)d_in[9];
  float* out = (float*)d_out;

  float* ws    = (float*)d_ws;
  float* ee    = ws;                    // B*S
  float* wts   = ee + BB * SS;          // B*S
  float* xh    = wts + BB * SS;         // B*KTOT
  float* cbuf  = xh + BB * KTOT;        // B*H
  float* gates = cbuf + BB * HH;        // B*G4
  float* bias  = gates + BB * G4;       // G4
  float* part  = bias + G4;             // SCH*B*H  (~2 MB)
  // total ~3.5 MB of d_ws

  enc_energy_kernel<<<BB * SS / 8, 256, 0, stream>>>(enc, attn_w, attn_b, ee);
  init_kernel<<<(BB * KTOT + 255) / 256, 256, 0, stream>>>(hidden, b_ih, b_hh, xh, cbuf, bias);

  for (int step = 0; step < STEPSN; ++step) {
    attn_softmax_kernel<<<BB, 256, 0, stream>>>(ee, attn_w, xh, wts);
    context_partial_kernel<<<dim3(SCH, BB), 128, 0, stream>>>(enc, wts, part);
    context_reduce_kernel<<<BB * HH / 4 / 256, 256, 0, stream>>>(part, xh);
    // gates GEMM: N=2048 -> 32 blocks x 512 threads (16 waves, 64x64 macro-tile)
    gemm_wmma_tiled<<<G4 / 64, 512, 0, stream>>>(xh, w_ih, KX, xh + KX, w_hh, HH,
                                                 KTOT, bias, gates, G4, nullptr, 0);
    lstm_kernel<<<(BB * HH) / 256, 256, 0, stream>>>(gates, cbuf, xh);
    // fc GEMM: N=256 -> 4 blocks; writes d_out[b, step, :] and next-step dec_in slot
    gemm_wmma_tiled<<<OUTN / 64, 512, 0, stream>>>(xh + KX, fc_w, HH, nullptr, nullptr, 0,
                                                   KTOT, fc_b, out + step * OUTN, STEPSN * OUTN,
                                                   xh, KTOT);
  }
}